// HashEncoder_9912784519603
// MI455X (gfx1250) — compile-verified
//
#include <hip/hip_runtime.h>

typedef float v2f __attribute__((ext_vector_type(2)));

// Instant-NGP constants (match reference): D=3, L=16, C=2, H=16, LOG2=19
// Dense levels l=0..2 (res=16,32,64), hashed levels l=3..15 (hsize = 2^19).

__global__ __launch_bounds__(256) void hashenc_fwd(
    const float* __restrict__ in,    // [B,3]
    const float* __restrict__ emb,   // [7131219,2]
    float* __restrict__ out,         // [B,32]
    int nB)
{
    const int b = blockIdx.x * 256 + threadIdx.x;
    if (b >= nB) return;

    // inputs are read exactly once -> non-temporal loads
    const float x = (__builtin_nontemporal_load(in + 3 * b + 0) + 1.0f) * 0.5f;
    const float y = (__builtin_nontemporal_load(in + 3 * b + 1) + 1.0f) * 0.5f;
    const float z = (__builtin_nontemporal_load(in + 3 * b + 2) + 1.0f) * 0.5f;

    float* const o = out + (size_t)b * 32;

    unsigned offset = 0;  // becomes per-level compile-time constant after unroll
#pragma unroll
    for (int l = 0; l < 16; ++l) {
        const unsigned res = 16u << l;
        const float rf = (float)res;
        const float px = x * rf, py = y * rf, pz = z * rf;
        const float gx = floorf(px), gy = floorf(py), gz = floorf(pz);
        const float tx = px - gx, ty = py - gy, tz = pz - gz;
        const unsigned ix = (unsigned)gx, iy = (unsigned)gy, iz = (unsigned)gz;

        unsigned idx[8];
        unsigned hsize;
        if (l <= 2) {
            // dense level: direct strided index; coordinates bounded by res+1,
            // so s < 2*hsize and the reference's (s % hsize) == conditional subtract.
            const unsigned r1 = res + 1u;
            hsize = r1 * r1 * r1;
            const unsigned hx0 = ix,            hx1 = ix + 1u;
            const unsigned hy0 = iy * r1,       hy1 = hy0 + r1;
            const unsigned hz0 = iz * r1 * r1,  hz1 = hz0 + r1 * r1;
#pragma unroll
            for (int c = 0; c < 8; ++c) {
                unsigned s = ((c & 1) ? hx1 : hx0)
                           + ((c & 2) ? hy1 : hy0)
                           + ((c & 4) ? hz1 : hz0);
                idx[c] = (s >= hsize) ? (s - hsize) : s;
            }
        } else {
            // hashed level: xor of coord*prime, uint32 wraparound; hsize = 2^19
            hsize = 1u << 19;
            const unsigned P1 = 2654435761u, P2 = 805459861u;
            const unsigned hx0 = ix,        hx1 = ix + 1u;
            const unsigned hy0 = iy * P1,   hy1 = hy0 + P1;
            const unsigned hz0 = iz * P2,   hz1 = hz0 + P2;
#pragma unroll
            for (int c = 0; c < 8; ++c) {
                unsigned s = ((c & 1) ? hx1 : hx0)
                           ^ ((c & 2) ? hy1 : hy0)
                           ^ ((c & 4) ? hz1 : hz0);
                idx[c] = s & ((1u << 19) - 1u);
            }
        }

        // Issue all 8 corner gathers (global_load_b64) before consuming: 8-deep MLP.
        v2f v[8];
#pragma unroll
        for (int c = 0; c < 8; ++c)
            v[c] = *(const v2f*)(emb + 2u * (offset + idx[c]));

        // trilinear weights: w[c] = prod_d (bit ? frac : 1-frac)
        const float wx1 = tx, wx0 = 1.0f - tx;
        const float wy1 = ty, wy0 = 1.0f - ty;
        const float wz1 = tz, wz0 = 1.0f - tz;
        const float wyz[4] = { wy0 * wz0, wy1 * wz0, wy0 * wz1, wy1 * wz1 };

        float a0 = 0.0f, a1 = 0.0f;
#pragma unroll
        for (int c = 0; c < 8; ++c) {
            const float w = ((c & 1) ? wx1 : wx0) * wyz[c >> 1];
            a0 = fmaf(w, v[c].x, a0);
            a1 = fmaf(w, v[c].y, a1);
        }

        // streaming 64 MB output -> NT store so it doesn't evict the L2-resident table
        v2f r; r.x = a0; r.y = a1;
        __builtin_nontemporal_store(r, (v2f*)(o + 2 * l));

        offset += hsize;
    }
}

extern "C" void kernel_launch(void* const* d_in, const int* in_sizes, int n_in,
                              void* d_out, int out_size, void* d_ws, size_t ws_size,
                              hipStream_t stream) {
    const float* in  = (const float*)d_in[0];   // inputs     [B,3] f32
    const float* emb = (const float*)d_in[1];   // embeddings [T,2] f32
    float* out       = (float*)d_out;           // [B,32] f32
    const int nB = in_sizes[0] / 3;
    const int blocks = (nB + 255) / 256;
    hipLaunchKernelGGL(hashenc_fwd, dim3(blocks), dim3(256), 0, stream,
                       in, emb, out, nB);
}